// ConvTemporalGraphical_5626407158358
// MI455X (gfx1250) — compile-verified
//
#include <hip/hip_runtime.h>
#include <cstdint>
#include <cstddef>

// Fused ST-GCN "ConvTemporalGraphical":
//   Stage 1 (per joint v):  Y[nc,q,v] = sum_t x[nc,t,v] * T[v,t,q]
//   Stage 2 (per time  t):  Z[nc,t,w] = sum_v Y[nc,t,v] * A[t,v,w]
// Roofline: 34.4 GFLOP vs ~1.07 GB fused HBM traffic -> ~46 us at 23.3 TB/s.
// fp32 WMMA 16x16x4 keeps reference precision. x streams through a 2-deep
// async-to-LDS ring (32 KiB eighths) so the TDM/async engine overlaps HBM
// streaming with WMMA instead of serializing fill -> wait0 -> compute.

typedef float v2f __attribute__((ext_vector_type(2)));
typedef float v8f __attribute__((ext_vector_type(8)));

#define NC_TOTAL      32768              // N*C
#define NC_BLK        16                 // nc rows per workgroup (WMMA M dim)
#define TILE          4096               // 64*64 floats per nc row
#define EIGHTH_FLOATS (NC_BLK * 8 * 64)  // 8192 floats = 32 KiB (8 t-rows)
#define XR_FLOATS     (2 * EIGHTH_FLOATS)
#define YS_FLOATS     (NC_BLK * 16 * 64) // 16384 floats = 64 KiB
#define LDS_BYTES     ((XR_FLOATS + YS_FLOATS) * 4)   // 128 KiB

typedef __attribute__((address_space(1))) int* as1_i32p;
typedef __attribute__((address_space(3))) int* as3_i32p;

// ---- async global->LDS copy (CDNA5 path), with compilable fallback ----
__device__ __forceinline__ void async_copy_b32(const float* g, float* l) {
#if __has_builtin(__builtin_amdgcn_global_load_async_to_lds_b32)
  __builtin_amdgcn_global_load_async_to_lds_b32(
      (as1_i32p)(uintptr_t)g,
      (as3_i32p)(uint32_t)(uintptr_t)l,
      0, 0);
#else
  *l = *g;
#endif
}

template <int N>
__device__ __forceinline__ void wait_async_le() {
#if __has_builtin(__builtin_amdgcn_s_wait_asynccnt)
  __builtin_amdgcn_s_wait_asynccnt(N);
#else
  asm volatile("s_wait_asynccnt %0" :: "i"(N) : "memory");
#endif
}

// Issue one t-eighth (8 t-rows x 16 nc x 64 v) of x into ring buffer `buf`.
// Each wave copies its own nc row: 8 t * 2 half-rows of 32 dwords = 16 ops.
__device__ __forceinline__ void issue_eighth(const float* __restrict__ x,
                                             float* __restrict__ xr,
                                             int ncbase, int wav, int lane,
                                             int ke, int buf) {
  const int nc = wav;
  const float* grow = x + (size_t)(ncbase + nc) * TILE + ke * 8 * 64;
  float* lrow = xr + buf * EIGHTH_FLOATS + nc * 512;
#pragma unroll
  for (int s = 0; s < 16; ++s) {
    const int tloc = s >> 1;
    const int v = (s & 1) * 32 + lane;
    async_copy_b32(grow + tloc * 64 + v, lrow + tloc * 64 + (v ^ nc));
  }
}

__global__ __launch_bounds__(512)
void stgcn_fused_wmma(const float* __restrict__ x,
                      const float* __restrict__ Tm,   // (V,T,Q)
                      const float* __restrict__ Am,   // (T,V,W)
                      float* __restrict__ z) {
  extern __shared__ float lds[];
  float* xr = lds;               // ring: 2 x [nc][tloc][v^nc]  (2*8192)
  float* ys = lds + XR_FLOATS;   // [nc][ql][v^nc^ql] for one q-chunk (16384)

  const int tid    = threadIdx.x;
  const int lane   = tid & 31;
  const int wav    = tid >> 5;      // 0..15
  const int g      = lane >> 4;     // lane half: 0/1
  const int ln     = lane & 15;     // N-index / M-index within half
  const int ncbase = blockIdx.x * NC_BLK;

  // Stage-1 accumulators for ALL 4 q-chunks: acc[qc*4+vv], v in [4*wav..+3].
  v8f acc[16] = {};

  // Prime the ring: eighths 0 and 1 in flight.
  issue_eighth(x, xr, ncbase, wav, lane, 0, 0);
  issue_eighth(x, xr, ncbase, wav, lane, 1, 1);

  // ---------------- Stage 1: stream K (=t) in eighths of 8 ----------------
#pragma unroll 1
  for (int ke = 0; ke < 8; ++ke) {
    if (ke == 7) wait_async_le<0>();     // last eighth: nothing newer in flight
    else         wait_async_le<16>();    // allow next eighth to stay in flight
    __syncthreads();                     // all waves' fills for `ke` visible

    const int buf = ke & 1;
    const float* xb = xr + buf * EIGHTH_FLOATS;

#pragma unroll
    for (int su = 0; su < 2; ++su) {     // two K-steps of 4 per eighth
      const int tA   = ke * 8 + su * 4 + 2 * g;   // K rows this lane-half
      const int tloc = su * 4 + 2 * g;
      // A-operands (X from LDS), reused across the 4 q-chunks.
      v2f a[4];
#pragma unroll
      for (int vv = 0; vv < 4; ++vv) {
        const int v = wav * 4 + vv;
        const float* ap = xb + ln * 512 + tloc * 64 + (v ^ ln);
        a[vv].x = ap[0];                 // K = tA
        a[vv].y = ap[64];                // K = tA+1
      }
#pragma unroll
      for (int qc = 0; qc < 4; ++qc) {
#pragma unroll
        for (int vv = 0; vv < 4; ++vv) {
          const int v = wav * 4 + vv;
          const float* tp = Tm + (size_t)v * TILE + tA * 64 + qc * 16 + ln;
          v2f b;
          b.x = tp[0];                   // B-op: T[v, tA,   q]
          b.y = tp[64];                  // B-op: T[v, tA+1, q]
          acc[qc * 4 + vv] = __builtin_amdgcn_wmma_f32_16x16x4_f32(
              false, a[vv], false, b, (short)0, acc[qc * 4 + vv], false, false);
        }
      }
    }

    __syncthreads();                     // everyone done reading `buf`
    if (ke < 6) issue_eighth(x, xr, ncbase, wav, lane, ke + 2, buf);
  }

  // ---------------- Stage 2: per q-chunk, Y -> LDS -> Z -------------------
#pragma unroll 1
  for (int qc = 0; qc < 4; ++qc) {
    // Scatter Y fragments of this chunk: ys[nc*1024 + ql*64 + (v^nc^ql)]
#pragma unroll
    for (int vv = 0; vv < 4; ++vv) {
      const int v = wav * 4 + vv;
#pragma unroll
      for (int r = 0; r < 8; ++r) {
        const int nc = r + 8 * g;        // D row: M = r + 8*half
        ys[nc * 1024 + ln * 64 + (v ^ nc ^ ln)] = acc[qc * 4 + vv][r];
      }
    }
    __syncthreads();

    // Z[nc, w] += Y[nc,v] * A[t,v,w]; one t-plane per wave.
    const int ql = wav;
    const int t  = qc * 16 + ql;
    v8f acc2[4] = {};
#pragma unroll 1
    for (int s = 0; s < 16; ++s) {
      const int vA = 4 * s + 2 * g;      // K rows this lane-half
      v2f a;
      const float* ybase = ys + ln * 1024 + ql * 64;
      a.x = ybase[(vA ^ ln ^ ql)];
      a.y = ybase[((vA + 1) ^ ln ^ ql)];
      const float* abase = Am + (size_t)t * TILE + vA * 64 + ln;
#pragma unroll
      for (int wt = 0; wt < 4; ++wt) {
        v2f b;
        b.x = abase[wt * 16];            // K = vA,   w = wt*16+ln
        b.y = abase[64 + wt * 16];       // K = vA+1
        acc2[wt] = __builtin_amdgcn_wmma_f32_16x16x4_f32(
            false, a, false, b, (short)0, acc2[wt], false, false);
      }
    }

    // Store Z fragments straight to HBM (64B per half-wave, fire-and-forget)
#pragma unroll
    for (int wt = 0; wt < 4; ++wt) {
#pragma unroll
      for (int r = 0; r < 8; ++r) {
        const int nc = r + 8 * g;
        z[(size_t)(ncbase + nc) * TILE + t * 64 + wt * 16 + ln] = acc2[wt][r];
      }
    }
    __syncthreads();                     // ys stable until all waves finish
  }
}

extern "C" void kernel_launch(void* const* d_in, const int* in_sizes, int n_in,
                              void* d_out, int out_size, void* d_ws, size_t ws_size,
                              hipStream_t stream) {
  (void)in_sizes; (void)n_in; (void)out_size; (void)d_ws; (void)ws_size;
  const float* x  = (const float*)d_in[0];
  const float* Tm = (const float*)d_in[1];
  const float* Am = (const float*)d_in[2];
  float* z = (float*)d_out;

  (void)hipFuncSetAttribute(reinterpret_cast<const void*>(stgcn_fused_wmma),
                            hipFuncAttributeMaxDynamicSharedMemorySize,
                            LDS_BYTES);

  const int grid = NC_TOTAL / NC_BLK;   // 2048 workgroups
  stgcn_fused_wmma<<<grid, 512, LDS_BYTES, stream>>>(x, Tm, Am, z);
}